// SimpleEdgePredictor_83786222011213
// MI455X (gfx1250) — compile-verified
//
#include <hip/hip_runtime.h>
#include <hip/hip_bf16.h>

// ---------------------------------------------------------------------------
// SimpleEdgePredictor on MI455X (gfx1250, wave32, WMMA).
//
// feat @ W1 decomposed:  h_i@W1a + h_j@W1b + smear@W1s + type*w320 + b1
//   - node projections (h@W1a / h@W1b) precomputed once per node into d_ws
//   - per-edge work: smear GEMM (K=64) + LN + ReLU + GEMM2 (256->8)
// All matmuls use V_WMMA_F32_16X16X4_F32 (fp32, matches fp32 reference).
// Weights are pre-packed in row-pairs so every WMMA B-fragment is ONE b64.
// ---------------------------------------------------------------------------

typedef __attribute__((ext_vector_type(2))) float v2f;
typedef __attribute__((ext_vector_type(8))) float v8f;

#define HID   128
#define HID2  256
#define MAXF  16
#define NG    64
#define NET   8
#define NGR   256    // B (number of graphs)
#define NMOL  64

static __device__ __forceinline__ v8f wmma4(v2f a, v2f b, v8f c) {
  // D = A(16x4) * B(4x16) + C(16x16), fp32
  return __builtin_amdgcn_wmma_f32_16x16x4_f32(
      /*neg_a=*/false, a, /*neg_b=*/false, b,
      /*c_mod=*/(short)0, c, /*reuse_a=*/false, /*reuse_b=*/false);
}

// ---------------------------------------------------------------------------
// Kernel 0: pack weights into b64-friendly pair-interleaved layouts.
//   Wpab[m][c][2]  = W1[2m+i][c],      m in [0,128)  (rows 0..255)
//   Wp1s[m][c][2]  = W1[256+2m+i][c],  m in [0,32)   (smear rows)
//   W2p [m][c][2]  = W2[2m+i][c] (c<8) else 0, c in [0,16), m in [0,128)
//   lnp [c][2]     = {ln_g[c], ln_b[c]}
//   cbp [c][2]     = {b1[c], b1[c]+W1[320][c]}   (ff / mf column bias)
// ---------------------------------------------------------------------------
__global__ __launch_bounds__(256) void sep_pack_kernel(
    const float* __restrict__ W1, const float* __restrict__ W2,
    const float* __restrict__ b1,
    const float* __restrict__ ln_g, const float* __restrict__ ln_b,
    float* __restrict__ Wpab, float* __restrict__ Wp1s,
    float* __restrict__ W2p, float* __restrict__ lnp,
    float* __restrict__ cbp) {
  const int t = blockIdx.x * 256 + threadIdx.x;
  const int N0 = 128 * 256;           // Wpab elements (pairs)
  const int N1 = N0 + 32 * 256;       // + Wp1s
  const int N2 = N1 + 128 * 16;       // + W2p
  const int N3 = N2 + 256;            // + lnp/cbp
  if (t < N0) {
    const int m = t >> 8, c = t & 255;
    Wpab[(m * 256 + c) * 2 + 0] = W1[(2 * m + 0) * HID2 + c];
    Wpab[(m * 256 + c) * 2 + 1] = W1[(2 * m + 1) * HID2 + c];
  } else if (t < N1) {
    const int u = t - N0, m = u >> 8, c = u & 255;
    Wp1s[(m * 256 + c) * 2 + 0] = W1[(256 + 2 * m + 0) * HID2 + c];
    Wp1s[(m * 256 + c) * 2 + 1] = W1[(256 + 2 * m + 1) * HID2 + c];
  } else if (t < N2) {
    const int u = t - N1, m = u >> 4, c = u & 15;
    W2p[(m * 16 + c) * 2 + 0] = (c < NET) ? W2[(2 * m + 0) * NET + c] : 0.0f;
    W2p[(m * 16 + c) * 2 + 1] = (c < NET) ? W2[(2 * m + 1) * NET + c] : 0.0f;
  } else if (t < N3) {
    const int c = t - N2;
    lnp[c * 2 + 0] = ln_g[c];
    lnp[c * 2 + 1] = ln_b[c];
    cbp[c * 2 + 0] = b1[c];
    cbp[c * 2 + 1] = b1[c] + W1[320 * HID2 + c];
  }
}

// ---------------------------------------------------------------------------
// Kernel 1: P[M,256] = H[M,128] @ W[128,256] using packed pair weights.
// One wave per 16x16 output tile; 32 WMMA per tile (K=128).
//   Wp points at the packed pair block for this 128-row weight slice.
// ---------------------------------------------------------------------------
__global__ __launch_bounds__(128) void sep_proj_kernel(
    const float* __restrict__ H, const float* __restrict__ Wp,
    float* __restrict__ P, int mtiles) {
  const int wave = threadIdx.x >> 5;
  const int lane = threadIdx.x & 31;
  const int rr = lane & 15;
  const int hi = lane >> 4;

  const int tile = blockIdx.x * 4 + wave;
  const int tm = tile >> 4;
  const int tn = tile & 15;
  if (tm >= mtiles) return;   // wave-uniform; EXEC stays all-ones

  const float* hrow = H + (size_t)(tm * 16 + rr) * HID;
  const int colB = tn * 16 + rr;
  const float* wcol = Wp + (size_t)colB * 2;   // + m*512 per pair-step

  v8f c = {};
#pragma unroll
  for (int k = 0; k < 32; ++k) {
    const int kk = 4 * k + 2 * hi;          // first K-row for this half-wave
    const int m = 2 * k + hi;               // packed pair index
    const v2f a = *(const v2f*)(hrow + kk);
    const v2f b = *(const v2f*)(wcol + (size_t)m * 512);
    c = wmma4(a, b, c);
  }
#pragma unroll
  for (int v = 0; v < 8; ++v)
    P[(size_t)(tm * 16 + v + 8 * hi) * HID2 + colB] = c[v];
}

// ---------------------------------------------------------------------------
// Kernel 2: fused edge MLP. One wave per 16-edge tile
// (graph b, source node i fixed; rows = the 16 frag targets j).
// ---------------------------------------------------------------------------
#define HSTRIDE 264   // 256 + 8 pad: dodge 64-bank conflicts, keep 8B align

__global__ __launch_bounds__(64) void sep_edge_kernel(
    const float* __restrict__ pos_mol, const float* __restrict__ pos_frag,
    const float* __restrict__ Pmol, const float* __restrict__ Pfi,
    const float* __restrict__ Pfj,
    const float* __restrict__ Wp1s, const float* __restrict__ W2p,
    const float* __restrict__ lnp, const float* __restrict__ cbp,
    const float* __restrict__ b2, float* __restrict__ out) {
  __shared__ float hbuf[2][16 * HSTRIDE];

  const int wave = threadIdx.x >> 5;
  const int lane = threadIdx.x & 31;
  const int rr = lane & 15;   // edge row j (and column-within-tile)
  const int hi = lane >> 4;

  const int tile = blockIdx.x * 2 + wave;
  const int FFT = NGR * MAXF;          // 4096 frag-frag tiles
  int bgraph, inode, is_mf;
  if (tile < FFT) { is_mf = 0; bgraph = tile >> 4; inode = tile & 15; }
  else { const int u = tile - FFT; is_mf = 1; bgraph = u >> 6; inode = u & 63; }

  // ---- distance for edge (i fixed, j = rr) ----
  const float* pi = is_mf ? (pos_mol + (size_t)(bgraph * NMOL + inode) * 3)
                          : (pos_frag + (size_t)(bgraph * MAXF + inode) * 3);
  const float* pj = pos_frag + (size_t)(bgraph * MAXF + rr) * 3;
  const float dx = pi[0] - pj[0], dy = pi[1] - pj[1], dz = pi[2] - pj[2];
  const float d = sqrtf(dx * dx + dy * dy + dz * dz + 1e-12f);

  // ---- Gaussian smearing: A-fragments for S[16,64] (16 K-steps) ----
  const float step = 10.0f / (float)(NG - 1);
  const float coeff = -0.5f / (step * step);
  v2f a[16];
#pragma unroll
  for (int k = 0; k < 16; ++k) {
    const int g = 4 * k + 2 * hi;
    const float t0 = d - (float)g * step;
    const float t1 = d - (float)(g + 1) * step;
    a[k].x = __expf(coeff * t0 * t0);
    a[k].y = __expf(coeff * t1 * t1);
  }

  // ---- GEMM1: S[16,64] @ W1s[64,256] -> 16 accumulators of 16x16 ----
  v8f acc[16];
  const v8f vzero = {};
#pragma unroll
  for (int n = 0; n < 16; ++n) acc[n] = vzero;
#pragma unroll
  for (int n = 0; n < 16; ++n) {
    const int col = n * 16 + rr;
    const float* wcol = Wp1s + (size_t)col * 2;
#pragma unroll
    for (int k = 0; k < 16; ++k) {
      const int m = 2 * k + hi;             // packed pair index (K rows 4k+2hi)
      const v2f bf = *(const v2f*)(wcol + (size_t)m * 512);
      acc[n] = wmma4(a[k], bf, acc[n]);
    }
  }

  // ---- add node projections + biases; accumulate LN statistics ----
  const float* Pi = is_mf ? (Pmol + (size_t)(bgraph * NMOL + inode) * HID2)
                          : (Pfi + (size_t)(bgraph * MAXF + inode) * HID2);
  const float* Pj = Pfj + (size_t)(bgraph * MAXF) * HID2;

  float s[8], q[8];
#pragma unroll
  for (int v = 0; v < 8; ++v) { s[v] = 0.0f; q[v] = 0.0f; }
#pragma unroll
  for (int n = 0; n < 16; ++n) {
    const int col = n * 16 + rr;
    const v2f cbv = *(const v2f*)(cbp + col * 2);   // {b1, b1+w320}
    const float cb = (is_mf ? cbv.y : cbv.x) + Pi[col];
#pragma unroll
    for (int v = 0; v < 8; ++v) {
      const float x = acc[n][v] + cb + Pj[(size_t)(v + 8 * hi) * HID2 + col];
      acc[n][v] = x;
      s[v] += x;
      q[v] += x * x;
    }
  }
  // reduce over the 16 lanes of each half-wave (rows v+8*hi stay lane-local)
#pragma unroll
  for (int m = 1; m <= 8; m <<= 1) {
#pragma unroll
    for (int v = 0; v < 8; ++v) {
      s[v] += __shfl_xor(s[v], m, 32);
      q[v] += __shfl_xor(q[v], m, 32);
    }
  }
  float mu[8], rs[8];
#pragma unroll
  for (int v = 0; v < 8; ++v) {
    mu[v] = s[v] * (1.0f / HID2);
    const float var = q[v] * (1.0f / HID2) - mu[v] * mu[v];
    rs[v] = rsqrtf(var + 1e-5f);
  }

  // ---- normalize + ReLU, stage h[16,256] into LDS ----
  float* hb = hbuf[wave];
#pragma unroll
  for (int n = 0; n < 16; ++n) {
    const int col = n * 16 + rr;
    const v2f ln = *(const v2f*)(lnp + col * 2);    // {gamma, beta}
#pragma unroll
    for (int v = 0; v < 8; ++v) {
      const float x = (acc[n][v] - mu[v]) * rs[v] * ln.x + ln.y;
      hb[(v + 8 * hi) * HSTRIDE + col] = fmaxf(x, 0.0f);
    }
  }

  // ---- GEMM2: h[16,256] @ W2pad[256,16] (cols 8..15 zero), 64 K-steps ----
  v8f c2 = vzero;
  const float* w2col = W2p + (size_t)rr * 2;
#pragma unroll
  for (int k = 0; k < 64; ++k) {
    const int kk = 4 * k + 2 * hi;
    const int m = 2 * k + hi;
    const v2f a2 = *(const v2f*)(hb + rr * HSTRIDE + kk);
    const v2f bf = *(const v2f*)(w2col + (size_t)m * 32);
    c2 = wmma4(a2, bf, c2);
  }

  // ---- store: pred_ff then pred_mf, concatenated flat ----
  float* obase = is_mf
      ? out + (size_t)NGR * MAXF * MAXF * NET +
            (size_t)(bgraph * NMOL + inode) * MAXF * NET
      : out + (size_t)(bgraph * MAXF + inode) * MAXF * NET;
  if (rr < NET) {
    const float bo = b2[rr];
#pragma unroll
    for (int v = 0; v < 8; ++v) {
      const int m = v + 8 * hi;             // edge row j
      obase[m * NET + rr] = c2[v] + bo;
    }
  }
}

// ---------------------------------------------------------------------------
extern "C" void kernel_launch(void* const* d_in, const int* in_sizes, int n_in,
                              void* d_out, int out_size, void* d_ws, size_t ws_size,
                              hipStream_t stream) {
  const float* h_mol    = (const float*)d_in[0];
  const float* pos_mol  = (const float*)d_in[1];
  const float* h_frag   = (const float*)d_in[2];
  const float* pos_frag = (const float*)d_in[3];
  // d_in[4]/d_in[5] (batch indices) unused: nodes are contiguous per graph
  const float* W1   = (const float*)d_in[6];
  const float* b1   = (const float*)d_in[7];
  const float* ln_g = (const float*)d_in[8];
  const float* ln_b = (const float*)d_in[9];
  const float* W2   = (const float*)d_in[10];
  const float* b2   = (const float*)d_in[11];

  // workspace layout (fp32)
  float* Pmol = (float*)d_ws;                      // [16384, 256]
  float* Pfi  = Pmol + (size_t)NGR * NMOL * HID2;  // [ 4096, 256]
  float* Pfj  = Pfi + (size_t)NGR * MAXF * HID2;   // [ 4096, 256]
  float* Wpab = Pfj + (size_t)NGR * MAXF * HID2;   // [128*256*2]
  float* Wp1s = Wpab + 128 * 256 * 2;              // [ 32*256*2]
  float* W2p  = Wp1s + 32 * 256 * 2;               // [128* 16*2]
  float* lnp  = W2p + 128 * 16 * 2;                // [256*2]
  float* cbp  = lnp + 256 * 2;                     // [256*2]

  // 0) pack weights (43264 work-items)
  sep_pack_kernel<<<169, 256, 0, stream>>>(W1, W2, b1, ln_g, ln_b,
                                           Wpab, Wp1s, W2p, lnp, cbp);

  // 1) node projections: h@W1[0:128] and h_frag@W1[128:256]
  sep_proj_kernel<<<4096, 128, 0, stream>>>(h_mol, Wpab, Pmol, 1024);
  sep_proj_kernel<<<1024, 128, 0, stream>>>(h_frag, Wpab, Pfi, 256);
  sep_proj_kernel<<<1024, 128, 0, stream>>>(h_frag, Wpab + 64 * 256 * 2, Pfj, 256);

  // 2) fused edge MLP: 4096 ff + 16384 mf tiles, 2 wave-tiles per block
  sep_edge_kernel<<<(4096 + 16384) / 2, 64, 0, stream>>>(
      pos_mol, pos_frag, Pmol, Pfi, Pfj, Wp1s, W2p, lnp, cbp, b2,
      (float*)d_out);
}